// Pointnet2BackboneCls_42554535969437
// MI455X (gfx1250) — compile-verified
//
#include <hip/hip_runtime.h>
#include <hip/hip_bf16.h>

typedef float v2f __attribute__((ext_vector_type(2)));
typedef float v8f __attribute__((ext_vector_type(8)));

#define FPS_T 256

// ---------------------------------------------------------------------------
// Farthest point sampling: one block per batch. Per-thread min-dist arrays
// stay in registers (PPT floats); LDS tree-argmax with lowest-index
// tie-breaking (matches jnp.argmax first-occurrence). Writes the gathered
// centers (new_xyz) directly; the index list itself is dead in the reference.
// ---------------------------------------------------------------------------
template<int PPT>
__global__ void fps_kernel(const float* __restrict__ xyz, int N, int npoint,
                           float* __restrict__ new_xyz)
{
  const int b   = blockIdx.x;
  const int tid = threadIdx.x;
  const float* pts  = xyz + (size_t)b * N * 3;
  float*       nout = new_xyz + (size_t)b * npoint * 3;

  float d[PPT];
#pragma unroll
  for (int j = 0; j < PPT; ++j) d[j] = 1e10f;

  __shared__ float s_val[FPS_T];
  __shared__ int   s_idx[FPS_T];
  __shared__ float s_c[3];

  int far = 0;
  for (int it = 0; it < npoint; ++it) {
    if (tid == 0) {
      float cx = pts[far*3+0], cy = pts[far*3+1], cz = pts[far*3+2];
      s_c[0] = cx; s_c[1] = cy; s_c[2] = cz;
      nout[it*3+0] = cx; nout[it*3+1] = cy; nout[it*3+2] = cz;
    }
    __syncthreads();
    const float cx = s_c[0], cy = s_c[1], cz = s_c[2];
    float best = -1.0f; int bi = 0;
#pragma unroll
    for (int j = 0; j < PPT; ++j) {
      const int p = tid + j * FPS_T;
      float dx = pts[p*3+0] - cx;
      float dy = pts[p*3+1] - cy;
      float dz = pts[p*3+2] - cz;
      float dd = dx*dx + dy*dy + dz*dz;
      float m  = fminf(d[j], dd);
      d[j] = m;
      if (m > best) { best = m; bi = p; }   // strict > keeps lowest index
    }
    s_val[tid] = best; s_idx[tid] = bi;
    __syncthreads();
    for (int off = FPS_T/2; off > 0; off >>= 1) {
      if (tid < off) {
        float v = s_val[tid+off]; int i2 = s_idx[tid+off];
        if (v > s_val[tid] || (v == s_val[tid] && i2 < s_idx[tid])) {
          s_val[tid] = v; s_idx[tid] = i2;
        }
      }
      __syncthreads();
    }
    far = s_idx[0];  // read happens before next iteration's writes (barrier at loop top)
  }
}

// ---------------------------------------------------------------------------
// Ball query: one thread per (batch, center). Emits the nsample smallest
// in-range indices in ascending order, padded with the first hit — the
// reference's top_k(-key) + where(sel>=N, first) semantics.
// ---------------------------------------------------------------------------
__global__ void ballq_kernel(const float* __restrict__ xyz,
                             const float* __restrict__ centers,
                             int* __restrict__ idx,
                             int B, int N, int P, int nsample, float r2)
{
  int t = blockIdx.x * blockDim.x + threadIdx.x;
  if (t >= B * P) return;
  int b = t / P;
  const float* pts = xyz + (size_t)b * N * 3;
  float cx = centers[t*3+0], cy = centers[t*3+1], cz = centers[t*3+2];
  int* out = idx + (size_t)t * nsample;
  int cnt = 0, first = 0;
  for (int j = 0; j < N; ++j) {
    float dx = pts[j*3+0] - cx;
    float dy = pts[j*3+1] - cy;
    float dz = pts[j*3+2] - cz;
    if (dx*dx + dy*dy + dz*dz < r2) {
      if (cnt == 0) first = j;
      out[cnt++] = j;
      if (cnt == nsample) break;
    }
  }
  for (; cnt < nsample; ++cnt) out[cnt] = first;
}

// ---------------------------------------------------------------------------
// Repack weights (Cout x Cin) into a zero-padded copy with stride CinP
// (CinP = Cin rounded up to a multiple of 4) so the WMMA B-fragment loads
// are unguarded, 8-byte-aligned global_load_b64.
// ---------------------------------------------------------------------------
__global__ void padw_kernel(const float* __restrict__ W, float* __restrict__ Wp,
                            int Cin, int CinP, int total)
{
  int t = blockIdx.x * blockDim.x + threadIdx.x;
  if (t >= total) return;
  int c = t / CinP, k = t - c * CinP;
  Wp[t] = (k < Cin) ? W[(size_t)c * Cin + k] : 0.0f;
}

// ---------------------------------------------------------------------------
// Shared-MLP layer on the f32 WMMA pipe (V_WMMA_F32_16X16X4_F32).
//   MODE 0: A from dense buffer X[row][k]          (Cin % 4 == 0, vector loads)
//   MODE 1: A gathered: k<3 -> (xyz[idx]-center)/radius, else feats[idx]
//   MODE 2: group-all concat: k<3 -> xyz[row], else feats[row]
// Each wave owns one 16-row M-tile and NT=4 column tiles (16x64 strip):
// per K-block one A v2f load feeds 4 WMMAs. Gather modes precompute the 3
// normalized xyz channels in a prologue (divides hoisted out of the loop)
// and only the final K-block carries bounds guards.
// Epilogue: relu(x*gamma+beta); `pool` max-reduces over the sample dim via
// atomicMax on float bits (valid: relu >= 0, pooled buffer pre-zeroed).
// ---------------------------------------------------------------------------
#define NT  4      // column tiles per wave
#define WPB 8      // waves per block (256 threads)

template<int MODE>
__global__ void __launch_bounds__(256)
mlp_layer_kernel(
    const float* __restrict__ X, int xStride,
    const float* __restrict__ xyz, const float* __restrict__ centers,
    const int*   __restrict__ idxT, const float* __restrict__ featsIn,
    int N, int P, int S, int Cf, float radius,
    const float* __restrict__ W, int wStride,
    const float* __restrict__ gamma, const float* __restrict__ beta,
    int Cin, int Cout,
    float* __restrict__ Y, int yStride, int pool)
{
  const int lane  = threadIdx.x & 31;
  const int mTile = blockIdx.x * WPB + (threadIdx.x >> 5);
  const int half  = lane >> 4;            // K-half select for A/B fragments
  const int l15   = lane & 15;
  const int row   = mTile * 16 + l15;     // A row (lanes 0-15 == lanes 16-31)
  const int colBase = blockIdx.y * (16 * NT);

  const float* wB[NT];
#pragma unroll
  for (int t = 0; t < NT; ++t)
    wB[t] = W + (size_t)(colBase + t * 16 + l15) * wStride;

  v8f acc[NT] = {};

  auto mm = [&](v2f a, int k0) {
#pragma unroll
    for (int t = 0; t < NT; ++t) {
      v2f bv = *(const v2f*)(wB[t] + k0 + half * 2);
      acc[t] = __builtin_amdgcn_wmma_f32_16x16x4_f32(
          false, a, false, bv, (short)0, acc[t], false, false);
    }
  };

  if constexpr (MODE == 0) {
    // dense: Cin is a multiple of 4; rows have even stride -> aligned v2f
    const float* xr = X + (size_t)row * xStride;
#pragma unroll 2
    for (int k0 = 0; k0 < Cin; k0 += 4) {
      v2f a = *(const v2f*)(xr + k0 + half * 2);
      mm(a, k0);
    }
  } else {
    // gather prologue: normalized/raw xyz channels + feature base pointer
    float q0, q1, q2;
    const float* fbase;
    if constexpr (MODE == 1) {
      int ps = row / S;                // b*P + p
      int b  = ps / P;
      int j  = idxT[row];
      const float* pj = xyz + ((size_t)b * N + j) * 3;
      const float* cc = centers + (size_t)ps * 3;
      q0 = (pj[0] - cc[0]) / radius;   // division matches reference rounding
      q1 = (pj[1] - cc[1]) / radius;
      q2 = (pj[2] - cc[2]) / radius;
      fbase = featsIn + ((size_t)b * N + j) * Cf;
    } else {
      const float* pj = xyz + (size_t)row * 3;
      q0 = pj[0]; q1 = pj[1]; q2 = pj[2];
      fbase = featsIn + (size_t)row * Cf;
    }
    const int CinP = (Cin + 3) & ~3;
    // K-block 0 covers k = 0..3 : (q0,q1 | q2,feat0) branchless per half
    {
      float f0 = (Cin > 3) ? fbase[0] : 0.0f;
      v2f a;
      a.x = half ? q2 : q0;
      a.y = half ? f0 : q1;
      mm(a, 0);
    }
    // full K-blocks: unguarded scalar feature loads (fbase[k-3], k odd-based)
    int k0 = 4;
#pragma unroll 2
    for (; k0 + 4 < CinP; k0 += 4) {
      const int ka = k0 + half * 2;
      v2f a;
      a.x = fbase[ka - 3];
      a.y = fbase[ka - 2];
      mm(a, k0);
    }
    // tail K-block (exists when CinP > 4; Cin % 4 == 3 -> one masked element)
    if (k0 < CinP) {
      const int ka = k0 + half * 2;
      v2f a;
      a.x = (ka     < Cin) ? fbase[ka - 3] : 0.0f;
      a.y = (ka + 1 < Cin) ? fbase[ka - 2] : 0.0f;
      mm(a, k0);
    }
  }

  // epilogue: relu(x*g+b), store or pooled atomic-max
#pragma unroll
  for (int t = 0; t < NT; ++t) {
    const int colg = colBase + t * 16 + l15;
    const float g  = gamma[colg];
    const float bb = beta[colg];
#pragma unroll
    for (int i = 0; i < 8; ++i) {
      // C/D layout: VGPR i -> rows i (lanes 0-15) and i+8 (lanes 16-31)
      const int grow = mTile * 16 + i + half * 8;
      float v = fmaxf(acc[t][i] * g + bb, 0.0f);
      if (pool) {
        atomicMax((unsigned int*)&Y[(size_t)(grow / S) * Cout + colg],
                  __float_as_uint(v));
      } else {
        Y[(size_t)grow * yStride + colg] = v;
      }
    }
  }
}

// ---------------------------------------------------------------------------
// Host side. Input order (setup_inputs flattened):
//   d_in[0]         pointcloud (8,16384,3) f32
//   d_in[1..9]      sa1: (w,g,b) x3 : [3->64->64->128]
//   d_in[10..18]    sa2: (w,g,b) x3 : [131->128->128->256]
//   d_in[19..27]    sa3: (w,g,b) x3 : [259->256->512->1024]
// d_out: (8,1024,1) f32 = 8192 floats.
// Workspace budget: ~71.5 MB (two 33.5 MB activation buffers + pools + pads).
// ---------------------------------------------------------------------------
extern "C" void kernel_launch(void* const* d_in, const int* in_sizes, int n_in,
                              void* d_out, int out_size, void* d_ws, size_t ws_size,
                              hipStream_t stream)
{
  (void)in_sizes; (void)n_in; (void)ws_size;
  const float* pc = (const float*)d_in[0];
  auto PRM = [&](int i) { return (const float*)d_in[i]; };

  char* ws = (char*)d_ws;
  auto carve = [&](size_t bytes) {
    char* p = ws;
    ws += (bytes + 255) & ~(size_t)255;
    return p;
  };
  float* bufA   = (float*)carve((size_t)8388608 * 4);       // 33.5 MB
  float* bufB   = (float*)carve((size_t)8388608 * 4);       // 33.5 MB
  float* feats1 = (float*)carve((size_t)8*512*128 * 4);     // 2 MB
  float* feats2 = (float*)carve((size_t)8*128*256 * 4);     // 1 MB
  float* nxyz1  = (float*)carve((size_t)8*512*3 * 4);
  float* nxyz2  = (float*)carve((size_t)8*128*3 * 4);
  int*   idx1   = (int*)  carve((size_t)8*512*32 * 4);
  int*   idx2   = (int*)  carve((size_t)8*128*64 * 4);
  float* w1p    = (float*)carve((size_t)64  * 4   * 4);     // sa1 L1: 64x4
  float* w2p    = (float*)carve((size_t)128 * 132 * 4);     // sa2 L1: 128x132
  float* w3p    = (float*)carve((size_t)256 * 260 * 4);     // sa3 L1: 256x260
  float* outF   = (float*)d_out;

  // pooled buffers accumulate via atomicMax -> must be zeroed every call
  hipMemsetAsync(feats1, 0, (size_t)8*512*128*4, stream);
  hipMemsetAsync(feats2, 0, (size_t)8*128*256*4, stream);
  hipMemsetAsync(outF,   0, (size_t)out_size * 4, stream);

  // zero-pad the odd-K gather-layer weights to 4-aligned K strides
  padw_kernel<<<(64*4    + 255)/256, 256, 0, stream>>>(PRM(1),  w1p, 3,   4,   64*4);
  padw_kernel<<<(128*132 + 255)/256, 256, 0, stream>>>(PRM(10), w2p, 131, 132, 128*132);
  padw_kernel<<<(256*260 + 255)/256, 256, 0, stream>>>(PRM(19), w3p, 259, 260, 256*260);

  const dim3 blk(256);   // 8 waves/block, one 16x64 output strip per wave

  // ---------------- SA1: N=16384 -> 512 centers, r=0.2, ns=32 ----------------
  fps_kernel<64><<<8, FPS_T, 0, stream>>>(pc, 16384, 512, nxyz1);
  ballq_kernel<<<(8*512 + 127)/128, 128, 0, stream>>>(
      pc, nxyz1, idx1, 8, 16384, 512, 32, (float)(0.2 * 0.2));

  // L1: gathered 3 -> 64     (R = 8*512*32 = 131072 rows, 8192 M-tiles)
  mlp_layer_kernel<1><<<dim3(8192/WPB, 64/64), blk, 0, stream>>>(
      nullptr, 0, pc, nxyz1, idx1, nullptr, 16384, 512, 32, 0, 0.2f,
      w1p, 4, PRM(2), PRM(3), 3, 64, bufA, 64, 0);
  // L2: 64 -> 64
  mlp_layer_kernel<0><<<dim3(8192/WPB, 64/64), blk, 0, stream>>>(
      bufA, 64, nullptr, nullptr, nullptr, nullptr, 0, 1, 32, 0, 1.0f,
      PRM(4), 64, PRM(5), PRM(6), 64, 64, bufB, 64, 0);
  // L3: 64 -> 128, max-pool over 32 samples -> feats1 (8,512,128)
  mlp_layer_kernel<0><<<dim3(8192/WPB, 128/64), blk, 0, stream>>>(
      bufB, 64, nullptr, nullptr, nullptr, nullptr, 0, 1, 32, 0, 1.0f,
      PRM(7), 64, PRM(8), PRM(9), 64, 128, feats1, 128, 1);

  // ---------------- SA2: N=512 -> 128 centers, r=0.4, ns=64 ------------------
  fps_kernel<2><<<8, FPS_T, 0, stream>>>(nxyz1, 512, 128, nxyz2);
  ballq_kernel<<<(8*128 + 127)/128, 128, 0, stream>>>(
      nxyz1, nxyz2, idx2, 8, 512, 128, 64, (float)(0.4 * 0.4));

  // L1: gathered 131 -> 128  (R = 8*128*64 = 65536 rows, 4096 M-tiles)
  mlp_layer_kernel<1><<<dim3(4096/WPB, 128/64), blk, 0, stream>>>(
      nullptr, 0, nxyz1, nxyz2, idx2, feats1, 512, 128, 64, 128, 0.4f,
      w2p, 132, PRM(11), PRM(12), 131, 128, bufA, 128, 0);
  // L2: 128 -> 128
  mlp_layer_kernel<0><<<dim3(4096/WPB, 128/64), blk, 0, stream>>>(
      bufA, 128, nullptr, nullptr, nullptr, nullptr, 0, 1, 64, 0, 1.0f,
      PRM(13), 128, PRM(14), PRM(15), 128, 128, bufB, 128, 0);
  // L3: 128 -> 256, max-pool over 64 samples -> feats2 (8,128,256)
  mlp_layer_kernel<0><<<dim3(4096/WPB, 256/64), blk, 0, stream>>>(
      bufB, 128, nullptr, nullptr, nullptr, nullptr, 0, 1, 64, 0, 1.0f,
      PRM(16), 128, PRM(17), PRM(18), 128, 256, feats2, 256, 1);

  // ---------------- SA3: group-all over 128 points ---------------------------
  // L1: concat(xyz2, feats2) 259 -> 256  (R = 8*128 = 1024 rows, 64 M-tiles)
  mlp_layer_kernel<2><<<dim3(64/WPB, 256/64), blk, 0, stream>>>(
      nullptr, 0, nxyz2, nullptr, nullptr, feats2, 0, 1, 128, 256, 1.0f,
      w3p, 260, PRM(20), PRM(21), 259, 256, bufA, 256, 0);
  // L2: 256 -> 512
  mlp_layer_kernel<0><<<dim3(64/WPB, 512/64), blk, 0, stream>>>(
      bufA, 256, nullptr, nullptr, nullptr, nullptr, 0, 1, 128, 0, 1.0f,
      PRM(22), 256, PRM(23), PRM(24), 256, 512, bufB, 512, 0);
  // L3: 512 -> 1024, max-pool over 128 points -> d_out (8,1024)
  mlp_layer_kernel<0><<<dim3(64/WPB, 1024/64), blk, 0, stream>>>(
      bufB, 512, nullptr, nullptr, nullptr, nullptr, 0, 1, 128, 0, 1.0f,
      PRM(25), 512, PRM(26), PRM(27), 512, 1024, outF, 1024, 1);
}